// pRNN_70342974373943
// MI455X (gfx1250) — compile-verified
//
#include <hip/hip_runtime.h>

typedef __attribute__((ext_vector_type(2))) float v2f;
typedef __attribute__((ext_vector_type(8))) float v8f;

#define L_   8
#define M_   128
#define K_   32
#define D_   16
#define B_   2048
#define N_   (L_ * M_)
#define BT   256                 // batch tile per workgroup
#define PSTR (2 * BT + 32)       // pair-row stride (dwords); ≡32 mod 64 -> half-waves hit disjoint banks
#define WSTR (K_ + 2)            // padded W row stride (even -> 8B-aligned b64 loads)

__global__ __launch_bounds__(256)
void prnn_layer_kernel(const float* __restrict__ x,
                       const float* __restrict__ trace,
                       const int*   __restrict__ src_node,
                       const int*   __restrict__ src_feat,
                       const float* __restrict__ W,
                       const float* __restrict__ bias,
                       float*       __restrict__ out,
                       int layer)
{
    // g stored pair-interleaved: gPair[p][b][j] = g[2p+j][b]  (p = k/2, j = k&1)
    __shared__ float gPair[(K_ / 2) * PSTR];   // 16 * 544 * 4B = 34 KB
    __shared__ float wLDS[D_ * WSTR];          // W[d][k], padded
    __shared__ float bLDS[D_];

    const int tid   = threadIdx.x;
    const int m     = blockIdx.x;            // node within layer
    const int btile = blockIdx.y;            // which 256-wide batch tile
    const int b0    = btile * BT;
    const int n     = layer * M_ + m;        // global node id

    // ---- stage W[n] (D x K) and bias[n] into LDS ----
    for (int i = tid; i < D_ * K_; i += 256) {
        int d = i >> 5, k = i & (K_ - 1);
        wLDS[d * WSTR + k] = W[(size_t)n * D_ * K_ + i];
    }
    if (tid < D_) bLDS[tid] = bias[n * D_ + tid];

    // ---- gather g[k][b] into LDS (pair-interleaved) ----
    // source row layout identical for trace/out: (node, B, D); x is (B, D)
    for (int k = 0; k < K_; ++k) {
        int s = src_node[n * K_ + k];        // uniform across block -> scalar
        int f = src_feat[n * K_ + k];
        const float* base;
        if (s == 0) {
            base = x + f;                                     // x[b, f]
        } else {
            int sl = (s - 1) >> 7;                            // source layer
            const float* arr = (sl < layer) ? out : trace;    // computed vs stale
            base = arr + (size_t)(s - 1) * (B_ * D_) + f;
        }
        gPair[(k >> 1) * PSTR + tid * 2 + (k & 1)] =
            base[(size_t)(b0 + tid) * D_];
    }
    __syncthreads();

    const int wave   = tid >> 5;
    const int lane   = tid & 31;
    const int hi     = lane >> 4;        // half-wave: K pairs {0,1} vs {2,3}
    const int lane16 = lane & 15;

    // ---- preload all 8 B-fragments (4x16 f32): B[k][d] = W[d][k] ----
    v2f bf[8];
#pragma unroll
    for (int ks = 0; ks < 8; ++ks) {
        int kb = 4 * ks + 2 * hi;                       // even
        bf[ks] = *(const v2f*)&wLDS[lane16 * WSTR + kb];  // ds_load_b64
    }
    const float bv = bLDS[lane16];

    // each wave computes 2 subtiles of 16 batch rows
#pragma unroll
    for (int sub = 0; sub < 2; ++sub) {
        const int bb = (wave * 2 + sub) * 16;   // local batch offset
        v8f acc = {0.f, 0.f, 0.f, 0.f, 0.f, 0.f, 0.f, 0.f};
#pragma unroll
        for (int ks = 0; ks < 8; ++ks) {
            const int p = 2 * ks + hi;          // pair row = (4*ks + 2*hi) / 2
            // A fragment: lanes hold rows b=bb+lane16, K-pair (2p, 2p+1)
            v2f af = *(const v2f*)&gPair[p * PSTR + (bb + lane16) * 2]; // ds_load_b64
            // D = A(16x4 f32) * B(4x16 f32) + C  -> v_wmma_f32_16x16x4_f32
            acc = __builtin_amdgcn_wmma_f32_16x16x4_f32(
                false, af, false, bf[ks], (short)0, acc, false, false);
        }
        // epilogue: bias + relu + store (two 64B contiguous segments per reg)
        const int d = lane16;
#pragma unroll
        for (int v = 0; v < 8; ++v) {
            int b = b0 + bb + v + hi * 8;
            float val = acc[v] + bv;
            val = val > 0.f ? val : 0.f;
            out[((size_t)n * B_ + b) * D_ + d] = val;
        }
    }
}

extern "C" void kernel_launch(void* const* d_in, const int* in_sizes, int n_in,
                              void* d_out, int out_size, void* d_ws, size_t ws_size,
                              hipStream_t stream) {
    (void)in_sizes; (void)n_in; (void)out_size; (void)d_ws; (void)ws_size;
    const float* x        = (const float*)d_in[0];
    const float* trace    = (const float*)d_in[1];
    const int*   src_node = (const int*)d_in[2];
    const int*   src_feat = (const int*)d_in[3];
    const float* W        = (const float*)d_in[4];
    const float* bias     = (const float*)d_in[5];
    float*       out      = (float*)d_out;

    dim3 grid(M_, B_ / BT);   // 128 nodes x 8 batch tiles
    dim3 block(256);
    for (int l = 0; l < L_; ++l) {
        prnn_layer_kernel<<<grid, block, 0, stream>>>(
            x, trace, src_node, src_feat, W, bias, out, l);
    }
}